// AttentionBlock_2954937499689
// MI455X (gfx1250) — compile-verified
//
#include <hip/hip_runtime.h>
#include <hip/hip_bf16.h>
#include <math.h>

// Problem constants (match reference)
#define BB 4
#define LL 512
#define HH 512
#define NHD 8
#define DKD 64          // HH / NHD
#define RIN 53
#define RHID 32

typedef float v2f __attribute__((ext_vector_type(2)));
typedef float v8f __attribute__((ext_vector_type(8)));

// ---------------------------------------------------------------------------
// fp32 WMMA: D = A(16x4) * B(4x16) + C   (exact fp32, CDNA5 matrix pipe)
// A frag: lane<16 -> row=lane, k={0,1}; lane>=16 -> row=lane-16, k={2,3}
// B frag: lane<16 -> col=lane, k={0,1}; lane>=16 -> col=lane-16, k={2,3}
// C/D   : vgpr i -> row=(lane<16 ? i : 8+i), col=lane%16
// ---------------------------------------------------------------------------
__device__ __forceinline__ v8f wmma4(v2f a, v2f b, v8f c) {
    return __builtin_amdgcn_wmma_f32_16x16x4_f32(false, a, false, b,
                                                 (short)0, c, false, false);
}

// CDNA5 async global->LDS copy (ASYNCcnt path), one b32 per lane.
__device__ __forceinline__ void async_ld_b32(void* lds, const void* gptr) {
    unsigned lo = (unsigned)(unsigned long long)(uintptr_t)lds;   // addrspace(3) offset
    unsigned long long ga = (unsigned long long)(uintptr_t)gptr;
    asm volatile("global_load_async_to_lds_b32 %0, %1, off"
                 :: "v"(lo), "v"(ga) : "memory");
}
__device__ __forceinline__ void async_wait0() {
    asm volatile("s_wait_asynccnt 0x0" ::: "memory");
}

// ===========================================================================
// K1 / K5: out = A[2048x512] @ W[512x512] + bias ; optional per-head transpose
// Block = 8 waves stacked on one 16-col strip (128 rows x 16 cols).
// The 512x16 weight strip is staged once in LDS via async loads, pair-
// interleaved so each B fragment is a single ds_load_b64.
// ===========================================================================
__global__ __launch_bounds__(256)
void gemm_bias_kernel(const float* __restrict__ A, const float* __restrict__ W,
                      const float* __restrict__ bias, float* __restrict__ out,
                      int headT) {
    __shared__ float s_w[HH * 16];          // 32 KB, pair-interleaved
    const int tn  = blockIdx.x & 31;        // 32 col strips
    const int tmg = blockIdx.x >> 5;        // 16 row groups of 128 rows
    const int t   = threadIdx.x;

    // stage W[:, tn*16 : tn*16+16] ; element (k,c) -> s_w[(k>>1)*32 + c*2 + (k&1)]
    for (int idx = t; idx < HH * 16; idx += 256) {
        const int k = idx >> 4, c = idx & 15;
        async_ld_b32(&s_w[(k >> 1) * 32 + (c << 1) + (k & 1)],
                     W + (size_t)k * HH + tn * 16 + c);
    }
    async_wait0();
    __syncthreads();

    const int wv   = t >> 5;
    const int lane = t & 31;
    const int half = lane >> 4;
    const int l16  = lane & 15;
    const int tm   = tmg * 8 + wv;          // this wave's 16-row tile

    const float* arow = A + (size_t)(tm * 16 + l16) * HH + half * 2;
    const float* swp  = &s_w[half * 32 + (l16 << 1)];

    v8f acc = {};
#pragma unroll 8
    for (int k0 = 0; k0 < HH; k0 += 4) {
        v2f a = *(const v2f*)(arow + k0);
        v2f b = *(const v2f*)(swp + (k0 >> 1) * 32);   // ds_load_b64
        acc = wmma4(a, b, acc);
    }

    const int colg = tn * 16 + l16;
    const float bv = bias[colg];
#pragma unroll
    for (int i = 0; i < 8; ++i) {
        const int rowg = tm * 16 + half * 8 + i;
        const float val = acc[i] + bv;
        if (headT) {   // [BL,H] -> [B,NH,L,DK]
            const int b_ = rowg >> 9, l_ = rowg & (LL - 1);
            const int h_ = colg >> 6, d_ = colg & (DKD - 1);
            out[(((size_t)b_ * NHD + h_) * LL + l_) * DKD + d_] = val;
        } else {
            out[(size_t)rowg * HH + colg] = val;
        }
    }
}

// ===========================================================================
// K2: scores[b,h,i,j] = (q . k) / sqrt(DK), 32 batched 512x512x64 GEMMs.
// Both fragments are contiguous float2 global loads (L2-resident q/k).
// ===========================================================================
__global__ __launch_bounds__(256)
void qk_scores_kernel(const float* __restrict__ q, const float* __restrict__ k,
                      float* __restrict__ scores) {
    const int wid  = blockIdx.x * 8 + (threadIdx.x >> 5);
    const int lane = threadIdx.x & 31;
    const int half = lane >> 4;
    const int l16  = lane & 15;
    const int batch = wid >> 10;            // 32*32 tiles per batch
    const int rem   = wid & 1023;
    const int tm = rem >> 5, tn = rem & 31;

    const size_t base = (size_t)batch * LL * DKD;
    const float* ar = q + base + (size_t)(tm * 16 + l16) * DKD + half * 2;
    const float* br = k + base + (size_t)(tn * 16 + l16) * DKD + half * 2;

    v8f acc = {};
#pragma unroll
    for (int k0 = 0; k0 < DKD; k0 += 4) {
        v2f a = *(const v2f*)(ar + k0);
        v2f b = *(const v2f*)(br + k0);
        acc = wmma4(a, b, acc);
    }

    float* srow = scores + (size_t)batch * LL * LL;
    const int colg = tn * 16 + l16;
#pragma unroll
    for (int i = 0; i < 8; ++i) {
        const int rowg = tm * 16 + half * 8 + i;
        srow[(size_t)rowg * LL + colg] = acc[i] * 0.125f;  // 1/sqrt(64)
    }
}

// ===========================================================================
// K3: fused refCov MLP + masked softmax(scores) + masked softmax(ref)
//     + combine + renormalize. One block per (b, i) row; comb overwrites
//     scores in place. 128 threads = 4 waves; wave w handles heads w, w+4.
//     refCov tiles staged with async-to-LDS (no VGPR round trip).
// ===========================================================================
__global__ __launch_bounds__(128)
void ref_combine_kernel(float* __restrict__ scores,
                        const float* __restrict__ refCov,
                        const unsigned char* __restrict__ mask,
                        const float* __restrict__ r1w, const float* __restrict__ r1b,
                        const float* __restrict__ r2w, const float* __restrict__ r2b) {
    __shared__ float s_r1w[RIN * RHID];     // 6784 B
    __shared__ float s_r1b[RHID];
    __shared__ float s_r2w[RHID * NHD];     // 1 KB
    __shared__ float s_r2b[NHD];
    __shared__ unsigned char s_mask[LL];    // 512 B
    __shared__ float s_cov[128 * RIN];      // 27 KB staging tile
    __shared__ float s_ref[NHD * LL];       // 16 KB ref logits

    const int b = blockIdx.x >> 9;
    const int i = blockIdx.x & (LL - 1);
    const int t = threadIdx.x;

    for (int idx = t; idx < RIN * RHID; idx += 128) s_r1w[idx] = r1w[idx];
    for (int idx = t; idx < RHID * NHD; idx += 128) s_r2w[idx] = r2w[idx];
    if (t < RHID) s_r1b[t] = r1b[t];
    if (t < NHD)  s_r2b[t] = r2b[t];
    for (int idx = t; idx < LL; idx += 128) s_mask[idx] = mask[b * LL + idx];
    __syncthreads();

    const bool mask_i = mask[b * LL + i] != 0;
    const float* covrow = refCov + (((size_t)b * LL + i) * LL) * RIN;

    // ---- MLP over all j, in 4 tiles of 128 ----
    for (int tile = 0; tile < 4; ++tile) {
        const int jbase = tile * 128;
        for (int idx = t; idx < 128 * RIN; idx += 128)
            async_ld_b32(&s_cov[idx], covrow + (size_t)jbase * RIN + idx);
        async_wait0();
        __syncthreads();

        const int j = jbase + t;
        float hid[RHID];
#pragma unroll
        for (int hh = 0; hh < RHID; ++hh) hid[hh] = s_r1b[hh];
        for (int c = 0; c < RIN; ++c) {
            const float xv = s_cov[t * RIN + c];
#pragma unroll
            for (int hh = 0; hh < RHID; ++hh)
                hid[hh] = fmaf(xv, s_r1w[c * RHID + hh], hid[hh]);
        }
        float o[NHD];
#pragma unroll
        for (int h = 0; h < NHD; ++h) o[h] = s_r2b[h];
#pragma unroll
        for (int hh = 0; hh < RHID; ++hh) {
            const float hv = fmaxf(hid[hh], 0.0f);
#pragma unroll
            for (int h = 0; h < NHD; ++h)
                o[h] = fmaf(hv, s_r2w[hh * NHD + h], o[h]);
        }
#pragma unroll
        for (int h = 0; h < NHD; ++h) s_ref[h * LL + j] = o[h];
        __syncthreads();
    }

    // ---- per-head masked softmax + combine (wave32 per head) ----
    const int w   = t >> 5;
    const int lid = t & 31;
    const float NEGINF = -__builtin_inff();

    for (int h = w; h < NHD; h += 4) {
        float* srow = scores + (((size_t)b * NHD + h) * LL + i) * LL;
        float sv[16], rv[16];
        float smax = NEGINF, rmax = NEGINF;
#pragma unroll
        for (int tt = 0; tt < 16; ++tt) {
            const int j = lid + 32 * tt;
            const bool vj = mask_i && (s_mask[j] != 0);
            sv[tt] = vj ? srow[j] : NEGINF;
            rv[tt] = vj ? s_ref[h * LL + j] : NEGINF;
            smax = fmaxf(smax, sv[tt]);
            rmax = fmaxf(rmax, rv[tt]);
        }
#pragma unroll
        for (int off = 16; off > 0; off >>= 1) {
            smax = fmaxf(smax, __shfl_xor(smax, off, 32));
            rmax = fmaxf(rmax, __shfl_xor(rmax, off, 32));
        }
        const bool uniform = (smax == NEGINF);   // row fully masked
        if (uniform) {
#pragma unroll
            for (int tt = 0; tt < 16; ++tt) srow[lid + 32 * tt] = 1.0f / (float)LL;
            continue;
        }
        float ssum = 0.f, rsum = 0.f;
#pragma unroll
        for (int tt = 0; tt < 16; ++tt) {
            sv[tt] = (sv[tt] == NEGINF) ? 0.f : __expf(sv[tt] - smax);
            rv[tt] = (rv[tt] == NEGINF) ? 0.f : __expf(rv[tt] - rmax);
            ssum += sv[tt];
            rsum += rv[tt];
        }
#pragma unroll
        for (int off = 16; off > 0; off >>= 1) {
            ssum += __shfl_xor(ssum, off, 32);
            rsum += __shfl_xor(rsum, off, 32);
        }
        const float is = 1.f / ssum, ir = 1.f / rsum;
        float csum = 0.f;
#pragma unroll
        for (int tt = 0; tt < 16; ++tt) {
            sv[tt] = sv[tt] * is + rv[tt] * ir;   // comb (pre-norm)
            csum += sv[tt];
        }
#pragma unroll
        for (int off = 16; off > 0; off >>= 1) csum += __shfl_xor(csum, off, 32);
        const float ic = 1.f / csum;
#pragma unroll
        for (int tt = 0; tt < 16; ++tt) srow[lid + 32 * tt] = sv[tt] * ic;
    }
}

// ===========================================================================
// K4: out[b,i,h*64+d] = sum_j comb[b,h,i,j] * v[b,h,j,d]  (WMMA batched)
// Block = 8 waves stacked on one 16-col strip of one batch; the 512x16
// v-strip is staged in LDS via async loads (shared by all 8 waves).
// ===========================================================================
__global__ __launch_bounds__(256)
void av_kernel(const float* __restrict__ comb, const float* __restrict__ v,
               float* __restrict__ out) {
    __shared__ float s_v[LL * 16];          // 32 KB, pair-interleaved
    const int batch = blockIdx.x >> 4;      // 32 batches x 16 blocks
    const int rem   = blockIdx.x & 15;
    const int rg    = rem >> 2;             // row group (8 tiles each)
    const int tn    = rem & 3;              // 4 col strips of 16
    const int t     = threadIdx.x;

    const size_t cb = (size_t)batch * LL * LL;
    const size_t vb = (size_t)batch * LL * DKD;

    for (int idx = t; idx < LL * 16; idx += 256) {
        const int k = idx >> 4, c = idx & 15;
        async_ld_b32(&s_v[(k >> 1) * 32 + (c << 1) + (k & 1)],
                     v + vb + (size_t)k * DKD + tn * 16 + c);
    }
    async_wait0();
    __syncthreads();

    const int wv   = t >> 5;
    const int lane = t & 31;
    const int half = lane >> 4;
    const int l16  = lane & 15;
    const int tm   = rg * 8 + wv;

    const float* ar  = comb + cb + (size_t)(tm * 16 + l16) * LL + half * 2;
    const float* svp = &s_v[half * 32 + (l16 << 1)];

    v8f acc = {};
#pragma unroll 8
    for (int k0 = 0; k0 < LL; k0 += 4) {
        v2f a = *(const v2f*)(ar + k0);
        v2f b = *(const v2f*)(svp + (k0 >> 1) * 32);   // ds_load_b64
        acc = wmma4(a, b, acc);
    }

    const int bidx = batch >> 3, h = batch & 7;
    const int cl = tn * 16 + l16;           // local dk
#pragma unroll
    for (int i = 0; i < 8; ++i) {
        const int rowg = tm * 16 + half * 8 + i;
        out[((size_t)bidx * LL + rowg) * HH + h * DKD + cl] = acc[i];
    }
}

// ===========================================================================
// K6: LayerNorm(residual + proj) ; one row per block
// ===========================================================================
__global__ __launch_bounds__(256)
void ln_kernel(const float* __restrict__ x, const float* __restrict__ pj,
               const float* __restrict__ g, const float* __restrict__ be,
               float* __restrict__ out) {
    const int r = blockIdx.x;
    const int t = threadIdx.x;
    const float* xr = x + (size_t)r * HH;
    const float* pr = pj + (size_t)r * HH;
    const float v0 = xr[t] + pr[t];
    const float v1 = xr[t + 256] + pr[t + 256];
    float s = v0 + v1, s2 = v0 * v0 + v1 * v1;
#pragma unroll
    for (int off = 16; off > 0; off >>= 1) {
        s  += __shfl_xor(s, off, 32);
        s2 += __shfl_xor(s2, off, 32);
    }
    __shared__ float w1[8], w2[8];
    const int w = t >> 5, lid = t & 31;
    if (lid == 0) { w1[w] = s; w2[w] = s2; }
    __syncthreads();
    if (t == 0) {
        float a = 0.f, b2 = 0.f;
        for (int i = 0; i < 8; ++i) { a += w1[i]; b2 += w2[i]; }
        w1[0] = a; w2[0] = b2;
    }
    __syncthreads();
    const float mu  = w1[0] * (1.0f / HH);
    const float var = w2[0] * (1.0f / HH) - mu * mu;
    const float inv = rsqrtf(var + 1e-5f);
    out[(size_t)r * HH + t]       = (v0 - mu) * inv * g[t] + be[t];
    out[(size_t)r * HH + t + 256] = (v1 - mu) * inv * g[t + 256] + be[t + 256];
}

// ===========================================================================
extern "C" void kernel_launch(void* const* d_in, const int* in_sizes, int n_in,
                              void* d_out, int out_size, void* d_ws, size_t ws_size,
                              hipStream_t stream) {
    const float* x    = (const float*)d_in[0];
    const unsigned char* mask = (const unsigned char*)d_in[1];   // jax bool -> 1B
    const float* refCov = (const float*)d_in[2];
    const float* wq = (const float*)d_in[3];  const float* bq = (const float*)d_in[4];
    const float* wk = (const float*)d_in[5];  const float* bk = (const float*)d_in[6];
    const float* wv = (const float*)d_in[7];  const float* bv = (const float*)d_in[8];
    const float* wo = (const float*)d_in[9];  const float* bo = (const float*)d_in[10];
    const float* r1w = (const float*)d_in[11]; const float* r1b = (const float*)d_in[12];
    const float* r2w = (const float*)d_in[13]; const float* r2b = (const float*)d_in[14];
    const float* lng = (const float*)d_in[15]; const float* lnb = (const float*)d_in[16];

    float* ws = (float*)d_ws;
    const size_t BLH = (size_t)BB * LL * HH;             // 1M floats
    float* q  = ws;
    float* k  = q + BLH;
    float* v  = k + BLH;
    float* sc = v + BLH;                                 // scores, then comb
    float* ao = sc + (size_t)BB * NHD * LL * LL;         // attn output [B,L,H]
    float* pj = ao + BLH;                                // projection

    const dim3 blk256(256), blk128(128);

    // QKV projections (head-transposed outputs), matrix pipe + LDS weight strip
    gemm_bias_kernel<<<512, blk256, 0, stream>>>(x, wq, bq, q, 1);
    gemm_bias_kernel<<<512, blk256, 0, stream>>>(x, wk, bk, k, 1);
    gemm_bias_kernel<<<512, blk256, 0, stream>>>(x, wv, bv, v, 1);

    // scores = q k^T / sqrt(dk)
    qk_scores_kernel<<<4096, blk256, 0, stream>>>(q, k, sc);

    // fused refCov MLP + masked softmaxes + combine (in-place over scores)
    ref_combine_kernel<<<BB * LL, blk128, 0, stream>>>(sc, refCov, mask,
                                                       r1w, r1b, r2w, r2b);

    // out = comb @ v   (back to [B,L,H])
    av_kernel<<<512, blk256, 0, stream>>>(sc, v, ao);

    // output projection
    gemm_bias_kernel<<<512, blk256, 0, stream>>>(ao, wo, bo, pj, 0);

    // residual + LayerNorm
    ln_kernel<<<BB * LL, blk256, 0, stream>>>(x, pj, lng, lnb, (float*)d_out);
}